// Loss_contrast_33182917329298
// MI455X (gfx1250) — compile-verified
//
#include <hip/hip_runtime.h>
#include <hip/hip_bf16.h>
#include <math.h>

typedef __attribute__((ext_vector_type(2))) float v2f;
typedef __attribute__((ext_vector_type(8))) float v8f;

#define HWOUT   10
#define CROPN   100
#define TOPOFF  206          // (512-100)/2
#define IN_W    512
#define SLICES  192          // 64 batch * 3 channels
#define EPSV    1e-4f
#define TPITCH  114          // tmp row pitch (even -> aligned b64, conflict-free)

// d_ws layout in floats
#define WS_W    0                         // 10*100 = 1000 (resize weights)
#define WS_RES  1024                      // 3 tensors * 192 slices * 100 vals
#define WS_PD   (1024 + 3*SLICES*100)     // 192 per-slice D partials
#define WS_PC   (WS_PD + SLICES)          // 192 per-slice C partials

// ---------------------------------------------------------------------------
// Kernel 1: build antialiased bilinear resize weights W[10][100].
// scale = 0.1 -> triangle kernel radius 10, row-normalized (jax antialias=True)
// ---------------------------------------------------------------------------
__global__ void k_weights(float* __restrict__ ws) {
    int i = threadIdx.x;
    if (i < HWOUT) {
        float center = 10.0f * (float)i + 4.5f;   // (i+0.5)/scale - 0.5
        float sum = 0.0f;
        for (int j = 0; j < CROPN; ++j) {
            float x = fabsf(center - (float)j) * 0.1f;
            sum += fmaxf(0.0f, 1.0f - x);
        }
        float inv = 1.0f / sum;
        for (int j = 0; j < CROPN; ++j) {
            float x = fabsf(center - (float)j) * 0.1f;
            ws[WS_W + i * CROPN + j] = fmaxf(0.0f, 1.0f - x) * inv;
        }
    }
}

// ---------------------------------------------------------------------------
// Kernel 2: resized = W * crop * W^T via V_WMMA_F32_16X16X4_F32.
// One wave per (tensor, slice); 8 waves/block; 576 jobs total.
// GEMM1: tmp[16x112] = Wpad[16x100] * crop[100x112]  (N-tiles in pairs -> two
//        independent accumulators per k-step, no WMMA RAW serialization)
// GEMM2: out[16x16]  = tmp[16x100] * W^T[100x16]
// The per-lane W fragment afrag[k] is identical for GEMM1-A and GEMM2-B
// (16x4 A-layout == 4x16 B-layout lane-wise), loaded from LDS exactly once.
// Inner loops are fully divergence-free (padded W rows + cndmask-zeroed B).
// ---------------------------------------------------------------------------
__global__ __launch_bounds__(256) void k_resize(const float* __restrict__ HRa,
                                                const float* __restrict__ HRb,
                                                const float* __restrict__ SRf,
                                                float* __restrict__ ws) {
    __shared__ float Wl[16 * CROPN];            // 6400 B, rows 10..15 zero
    __shared__ float tmp[8][16 * TPITCH];       // 58368 B

    for (int i = threadIdx.x; i < 16 * CROPN; i += blockDim.x)
        Wl[i] = (i < HWOUT * CROPN) ? ws[WS_W + i] : 0.0f;
    __syncthreads();

    const int wave = threadIdx.x >> 5;
    const int lane = threadIdx.x & 31;
    const int gw   = blockIdx.x * 8 + wave;     // 0..575
    const int ten  = gw / SLICES;               // 0=A 1=B 2=SR
    const int s    = gw % SLICES;

    const float* src  = (ten == 0) ? HRa : ((ten == 1) ? HRb : SRf);
    const float* crop = src + (size_t)s * IN_W * IN_W + (size_t)TOPOFF * IN_W + TOPOFF;

    const int half = lane >> 4;                 // lanes 0-15: K+0/1, 16-31: K+2/3
    const int l16  = lane & 15;
    float* mytmp = &tmp[wave][0];

    // ---- hoist W fragments: afrag[k] = (W[l16][4k+2h], W[l16][4k+2h+1]) ----
    v2f afrag[25];
#pragma unroll
    for (int k = 0; k < 25; ++k) {
        const int off = l16 * CROPN + 4 * k + 2 * half;   // 8B-aligned -> ds_load_b64
        afrag[k].x = Wl[off];
        afrag[k].y = Wl[off + 1];
    }

    // ---- GEMM1: two N-tiles per pass ----
    for (int nt = 0; nt < 7; nt += 2) {
        v8f acc0 = {};
        v8f acc1 = {};
        const int  col0 = nt * 16 + l16;
        const int  col1 = col0 + 16;
        const bool ok0  = (col0 < CROPN);
        const bool ok1  = (col1 < CROPN);
        const int  cc0  = ok0 ? col0 : (CROPN - 1);       // clamp: loads stay uniform
        const int  cc1  = ok1 ? col1 : (CROPN - 1);
        for (int k = 0; k < 25; ++k) {
            const int r = 4 * k + 2 * half;
            const float* rp = crop + (size_t)r * IN_W;
            const float x0 = rp[cc0], y0 = rp[IN_W + cc0];
            const float x1 = rp[cc1], y1 = rp[IN_W + cc1];
            v2f b0, b1;
            b0.x = ok0 ? x0 : 0.0f;  b0.y = ok0 ? y0 : 0.0f;
            b1.x = ok1 ? x1 : 0.0f;  b1.y = ok1 ? y1 : 0.0f;
            acc0 = __builtin_amdgcn_wmma_f32_16x16x4_f32(false, afrag[k], false, b0,
                                                         (short)0, acc0, false, false);
            acc1 = __builtin_amdgcn_wmma_f32_16x16x4_f32(false, afrag[k], false, b1,
                                                         (short)0, acc1, false, false);
        }
        // D-tile: VGPR v, lanes 0-15 -> row v, lanes 16-31 -> row v+8
#pragma unroll
        for (int v = 0; v < 8; ++v)
            mytmp[(v + 8 * half) * TPITCH + nt * 16 + l16] = acc0[v];
        if (nt + 1 < 7) {
#pragma unroll
            for (int v = 0; v < 8; ++v)
                mytmp[(v + 8 * half) * TPITCH + (nt + 1) * 16 + l16] = acc1[v];
        }
    }

    // ---- GEMM2: A from tmp (aligned ds_load_b64), B = afrag (reused) ----
    v8f acc2 = {};
    for (int k = 0; k < 25; ++k) {
        const int kk = 4 * k + 2 * half;
        v2f a;
        a.x = mytmp[l16 * TPITCH + kk];
        a.y = mytmp[l16 * TPITCH + kk + 1];
        acc2 = __builtin_amdgcn_wmma_f32_16x16x4_f32(false, a, false, afrag[k],
                                                     (short)0, acc2, false, false);
    }

    float* out = ws + WS_RES + (size_t)(ten * SLICES + s) * 100;
#pragma unroll
    for (int v = 0; v < 8; ++v) {
        const int i = v + 8 * half;
        const int j = l16;
        if (i < HWOUT && j < HWOUT) out[i * HWOUT + j] = acc2[v];
    }
}

// ---------------------------------------------------------------------------
// Kernel 3: per-slice D partial and pairwise C partial.
// IAVG = (A+B)/2 exactly (crop+resize are linear), so no 4th resize needed.
// One block of 256 threads per slice; fixed-order tree reduction (deterministic).
// ---------------------------------------------------------------------------
__global__ __launch_bounds__(256) void k_pair(float* __restrict__ ws) {
    __shared__ float As[100], Bs[100], Ss[100];
    __shared__ float red[256];

    const int s = blockIdx.x;
    const int t = threadIdx.x;

    const float* RA = ws + WS_RES + (size_t)(0 * SLICES + s) * 100;
    const float* RB = ws + WS_RES + (size_t)(1 * SLICES + s) * 100;
    const float* RS = ws + WS_RES + (size_t)(2 * SLICES + s) * 100;
    if (t < 100) { As[t] = RA[t]; Bs[t] = RB[t]; Ss[t] = RS[t]; }
    __syncthreads();

    // D contribution
    float d = 0.0f;
    if (t < 100) {
        float e = Ss[t] - 0.5f * (As[t] + Bs[t]);
        d = e * e;
    }

    // C contribution over pairs (i != j)
    float csum = 0.0f;
    for (int p = t; p < 10000; p += 256) {
        const int i = p / 100, j = p % 100;
        if (i == j) continue;
        const float dr = (float)(i / 10) - (float)(j / 10);
        const float dc = (float)(i % 10) - (float)(j % 10);
        const float dpos = sqrtf(dr * dr + dc * dc);
        const float dint = 1.0f - tanhf(0.5f * (fabsf(As[i] - As[j]) + fabsf(Bs[i] - Bs[j])));
        csum += fabsf(Ss[i] - Ss[j]) / (dpos * dint + EPSV);
    }

    red[t] = d;
    __syncthreads();
    for (int off = 128; off > 0; off >>= 1) {
        if (t < off) red[t] += red[t + off];
        __syncthreads();
    }
    if (t == 0) ws[WS_PD + s] = red[0];
    __syncthreads();

    red[t] = csum;
    __syncthreads();
    for (int off = 128; off > 0; off >>= 1) {
        if (t < off) red[t] += red[t + off];
        __syncthreads();
    }
    if (t == 0) ws[WS_PC + s] = red[0];
}

// ---------------------------------------------------------------------------
// Kernel 4: fixed-order final sum, out = |D - C|
// ---------------------------------------------------------------------------
__global__ void k_final(const float* __restrict__ ws, float* __restrict__ out) {
    if (threadIdx.x == 0 && blockIdx.x == 0) {
        float D = 0.0f, C = 0.0f;
        for (int s = 0; s < SLICES; ++s) {
            D += ws[WS_PD + s];
            C += ws[WS_PC + s];
        }
        out[0] = fabsf(D - C);
    }
}

extern "C" void kernel_launch(void* const* d_in, const int* in_sizes, int n_in,
                              void* d_out, int out_size, void* d_ws, size_t ws_size,
                              hipStream_t stream) {
    const float* HRa = (const float*)d_in[0];
    const float* HRb = (const float*)d_in[1];
    const float* SRf = (const float*)d_in[2];
    float* ws  = (float*)d_ws;
    float* out = (float*)d_out;

    k_weights<<<1, 64, 0, stream>>>(ws);
    k_resize<<<72, 256, 0, stream>>>(HRa, HRb, SRf, ws);   // 576 waves, 1 job each
    k_pair<<<SLICES, 256, 0, stream>>>(ws);
    k_final<<<1, 32, 0, stream>>>(ws, out);
}